// DeletionChannel_9680856285943
// MI455X (gfx1250) — compile-verified
//
#include <hip/hip_runtime.h>
#include <stdint.h>

#define P_DEL 0.1f
#define B_    64
#define L_    4096
#define V_    128

typedef unsigned int v4u_t __attribute__((ext_vector_type(4)));
typedef int          v8i_t __attribute__((ext_vector_type(8)));
typedef int          v4i_t __attribute__((ext_vector_type(4)));

// ---------------------------------------------------------------------------
// Kernel 1: per-batch stable stream-compaction permutation.
// One block per batch (64 blocks x 1024 threads); each thread owns 4 positions.
// perm[b*L + d] = source row for output row d, or -1 for the eos-padded tail.
// ---------------------------------------------------------------------------
__global__ __launch_bounds__(1024) void build_perm_kernel(
    const float* __restrict__ noise, const int* __restrict__ msg_len,
    int* __restrict__ perm) {
  const int b   = blockIdx.x;
  const int tid = threadIdx.x;
  const int lim = msg_len[b] - 1;                 // positions < lim are deletable

  const float4 nv =
      reinterpret_cast<const float4*>(noise + (size_t)b * L_)[tid];
  const int base = tid << 2;
  const int k0 = !((nv.x < P_DEL) && (base + 0 < lim));
  const int k1 = !((nv.y < P_DEL) && (base + 1 < lim));
  const int k2 = !((nv.z < P_DEL) && (base + 2 < lim));
  const int k3 = !((nv.w < P_DEL) && (base + 3 < lim));
  const int tsum = k0 + k1 + k2 + k3;

  __shared__ int sbuf[1024];
  sbuf[tid] = tsum;
  __syncthreads();
  for (int off = 1; off < 1024; off <<= 1) {      // Hillis-Steele inclusive scan
    const int v   = sbuf[tid];
    const int add = (tid >= off) ? sbuf[tid - off] : 0;
    __syncthreads();
    sbuf[tid] = v + add;
    __syncthreads();
  }
  const int incl  = sbuf[tid];
  const int total = sbuf[1023];                   // number of kept rows

  int* p = perm + (size_t)b * L_;
  int  d = incl - tsum;                           // exclusive prefix
  if (k0) p[d++] = base;
  if (k1) p[d++] = base + 1;
  if (k2) p[d++] = base + 2;
  if (k3) p[d++] = base + 3;
  for (int i = total + tid; i < L_; i += 1024) p[i] = -1;  // eos tail marker
}

// ---------------------------------------------------------------------------
// Kernel 2: TDM gather-mode row gather (global -> LDS), then coalesced B128
// stores (LDS -> global). One wave per 16-row tile; indices are non-decreasing
// (kept rows ascending, then 0xFFFF tail), so TDM OOB-returns-zero is safe;
// eos rows are substituted at store time.
// ---------------------------------------------------------------------------
__global__ __launch_bounds__(32) void gather_tdm_kernel(
    const float* __restrict__ msg, const int* __restrict__ perm,
    float* __restrict__ out) {
  __shared__ float tile[16 * V_];                 // 8 KB per wave

  const int tileIdx = blockIdx.x;                 // B*(L/16) = 16384 tiles
  const int b       = tileIdx >> 8;               // 256 tiles per batch
  const int row0    = (tileIdx & 255) << 4;

  const int* p = perm + ((size_t)b * L_ + row0);
  int idx[16];
#pragma unroll
  for (int i = 0; i < 16; ++i)
    idx[i] = __builtin_amdgcn_readfirstlane(p[i]);  // force SGPR for D#

  const uint64_t gaddr  = (uint64_t)(uintptr_t)(msg + (size_t)b * L_ * V_);
  const uint32_t ldsoff = (uint32_t)(uintptr_t)(&tile[0]);  // LDS byte offset

  // D# group 0: count=1 | gather_mode=1 (16-bit indices), lds_addr,
  // 57-bit global tile base, type=2 ("image") in bits [127:126].
  v4u_t g0;
  g0[0] = 0x80000001u;
  g0[1] = ldsoff;
  g0[2] = (uint32_t)gaddr;
  g0[3] = ((uint32_t)(gaddr >> 32) & 0x01FFFFFFu) | (2u << 30);

  // D# group 1: wg_mask=0, data_size=2 (4 B), tensor_dim0=128,
  // tensor_dim1=4096 (row OOB bound), tile_dim0=128, tile_dim1=16 indices,
  // tensor_dim0_stride=128 elements. dim1_stride/tile_dim2 ignored in gather.
  v8i_t g1;
  g1[0] = (int)(2u << 16);
  g1[1] = (int)((uint32_t)V_ << 16);   // tensor_dim0[15:0] at bits 63:48
  g1[2] = (int)((uint32_t)L_ << 16);   // tensor_dim1[15:0] at bits 95:80
  g1[3] = (int)((uint32_t)V_ << 16);   // tile_dim0 at bits 127:112
  g1[4] = 16;                          // tile_dim1 = #valid indices
  g1[5] = V_;                          // tensor_dim0_stride[31:0]
  g1[6] = 0;
  g1[7] = 0;

  // D# groups 2/3: sixteen packed 16-bit row indices (-1 -> 0xFFFF = OOB).
  v4i_t g2, g3;
#pragma unroll
  for (int i = 0; i < 4; ++i)
    g2[i] = (idx[2 * i] & 0xFFFF) | ((idx[2 * i + 1] & 0xFFFF) << 16);
#pragma unroll
  for (int i = 0; i < 4; ++i)
    g3[i] = (idx[8 + 2 * i] & 0xFFFF) | ((idx[8 + 2 * i + 1] & 0xFFFF) << 16);

  // 6-arg form on this toolchain (clang-23 / therock headers): the extra
  // v8i group is zero-filled (matches the probe's verified zero call).
  const v8i_t gx = {0, 0, 0, 0, 0, 0, 0, 0};
  __builtin_amdgcn_tensor_load_to_lds(g0, g1, g2, g3, gx, 0);
  __builtin_amdgcn_s_wait_tensorcnt(0);
  asm volatile("" ::: "memory");                  // LDS now holds the tile

  const int     lane = threadIdx.x;
  float4*       o4 = reinterpret_cast<float4*>(out + ((size_t)b * L_ + row0) * V_);
  const float4* t4 = reinterpret_cast<const float4*>(tile);
#pragma unroll
  for (int r = 0; r < 16; ++r) {
    float4 v = t4[r * 32 + lane];
    if (idx[r] < 0)                               // eos one-hot row
      v = make_float4(lane == 0 ? 1.0f : 0.0f, 0.0f, 0.0f, 0.0f);
    o4[r * 32 + lane] = v;                        // 512 B contiguous per row
  }
}

// ---------------------------------------------------------------------------
extern "C" void kernel_launch(void* const* d_in, const int* in_sizes, int n_in,
                              void* d_out, int out_size, void* d_ws,
                              size_t ws_size, hipStream_t stream) {
  (void)in_sizes; (void)n_in; (void)out_size; (void)ws_size;
  const float* msg   = (const float*)d_in[0];
  const float* noise = (const float*)d_in[1];
  const int*   mlen  = (const int*)d_in[2];
  float*       out   = (float*)d_out;
  int*         perm  = (int*)d_ws;                // B*L ints = 1 MiB scratch

  build_perm_kernel<<<B_, 1024, 0, stream>>>(noise, mlen, perm);
  gather_tdm_kernel<<<B_ * (L_ / 16), 32, 0, stream>>>(msg, perm, out);
}